// SimpleGNN_67430986547654
// MI455X (gfx1250) — compile-verified
//
#include <hip/hip_runtime.h>
#include <hip/hip_bf16.h>

typedef float v2f __attribute__((ext_vector_type(2)));
typedef float v8f __attribute__((ext_vector_type(8)));

// ---------------- degree / norm ----------------
__global__ void deg_init_kernel(float* deg, int n) {
    int i = blockIdx.x * blockDim.x + threadIdx.x;
    if (i < n) deg[i] = 1.0f;  // self-loop
}
__global__ void deg_count_kernel(const int* __restrict__ dst, float* deg, int e) {
    int i = blockIdx.x * blockDim.x + threadIdx.x;
    if (i < e) atomicAdd(&deg[dst[i]], 1.0f);
}
__global__ void make_dis_kernel(float* deg, int n) {
    int i = blockIdx.x * blockDim.x + threadIdx.x;
    if (i < n) {
        float d = deg[i];
        deg[i] = (d > 0.0f) ? rsqrtf(d) : 0.0f;
    }
}

// ---------------- WMMA GEMM with row-scale epilogue ----------------
// C[M,N] = rowscale(A[M,K] * B[K,N]); optionally duplicate into C2 (self-loop init).
// 8 waves/block; wave w owns rows [blk*128 + 16w, +16), all N columns in registers.
// B staged in LDS as k-pairs: lb[(p*N+n)*2 + {0,1}] = B[2p][n], B[2p+1][n]
// -> one b64 LDS read per lane per fragment, matching the 16x4 f32 A-layout
//    (lanes 0..15: k0,k0+1 ; lanes 16..31: k0+2,k0+3).
template <int K, int N>
__global__ __launch_bounds__(256) void gemm_rowscale_kernel(
    const float* __restrict__ A, const float* __restrict__ B,
    const float* __restrict__ rs, float* __restrict__ C, float* __restrict__ C2,
    int M) {
    constexpr int NT = N / 16;              // 16x16 tiles per wave
    __shared__ float lb[(K / 2) * N * 2];   // K*N floats, pair-interleaved

    const int tid = threadIdx.x;
    for (int idx = tid; idx < (K / 2) * N; idx += 256) {
        int p = idx / N, n = idx % N;
        lb[idx * 2 + 0] = B[(2 * p + 0) * N + n];
        lb[idx * 2 + 1] = B[(2 * p + 1) * N + n];
    }
    __syncthreads();

    const int wave = tid >> 5;
    const int lane = tid & 31;
    const int half = lane >> 4;   // 0: k0,k0+1   1: k0+2,k0+3
    const int l16  = lane & 15;

    const long row0 = (long)blockIdx.x * 128 + wave * 16;
    long arow = row0 + l16;                       // A-fragment row (M index)
    if (arow >= M) arow = (long)M - 1;            // clamp; stores are guarded
    const float* __restrict__ Arow = A + arow * K;

    v8f acc[NT] = {};

    for (int k0 = 0; k0 < K; k0 += 4) {
        const v2f af = *(const v2f*)(Arow + k0 + 2 * half);   // global b64
        const int p = (k0 >> 1) + half;
#pragma unroll
        for (int t = 0; t < NT; ++t) {
            const v2f bfr = *(const v2f*)(&lb[((long)p * N + t * 16 + l16) * 2]);  // ds b64
            acc[t] = __builtin_amdgcn_wmma_f32_16x16x4_f32(
                false, af, false, bfr, (short)0, acc[t], false, false);
        }
    }

    // C/D layout: reg r, lanes 0..15 -> M=r ; lanes 16..31 -> M=r+8 ; N = lane%16
#pragma unroll
    for (int r = 0; r < 8; ++r) {
        const long rowr = row0 + r + 8 * half;
        if (rowr < M) {
            const float sc = rs ? rs[rowr] : 1.0f;
#pragma unroll
            for (int t = 0; t < NT; ++t) {
                const float v = acc[t][r] * sc;
                const long o = rowr * N + t * 16 + l16;
                C[o] = v;
                if (C2) C2[o] = v;   // self-loop initialisation of the accumulator
            }
        }
    }
}

// ---------------- edge scatter-add (pure sum; norms already folded) ----------
template <int F>
__global__ __launch_bounds__(256) void scatter_add_kernel(
    const int* __restrict__ src, const int* __restrict__ dst,
    const float* __restrict__ msg, float* __restrict__ out, long e) {
    constexpr int C = F / 4;
    const long idx = (long)blockIdx.x * blockDim.x + threadIdx.x;
    const long eidx = idx / C;
    const int c = (int)(idx % C);
    if (eidx >= e) return;
    const int s = src[eidx];
    const int d = dst[eidx];
    const float4 v = *(const float4*)(msg + (long)s * F + c * 4);
    float* o = out + (long)d * F + c * 4;
    atomicAdd(o + 0, v.x);
    atomicAdd(o + 1, v.y);
    atomicAdd(o + 2, v.z);
    atomicAdd(o + 3, v.w);
}

// ---------------- epilogue: h = relu(dis[row]*h + b[col]) --------------------
template <int F>
__global__ void finish_layer_kernel(float* __restrict__ h,
                                    const float* __restrict__ dis,
                                    const float* __restrict__ b, long n) {
    const long idx = (long)blockIdx.x * blockDim.x + threadIdx.x;
    if (idx >= n * F) return;
    const long i = idx / F;
    const int f = (int)(idx % F);
    const float v = h[idx] * dis[i] + b[f];
    h[idx] = v > 0.0f ? v : 0.0f;
}

// ---------------- head: out = softmax(h @ Wf + bf), Wf 64x10 -----------------
__global__ __launch_bounds__(256) void head_softmax_kernel(
    const float* __restrict__ h, const float* __restrict__ Wf,
    const float* __restrict__ bfv, float* __restrict__ out, int n) {
    __shared__ float w[64 * 10];
    __shared__ float bb[10];
    for (int i = threadIdx.x; i < 640; i += blockDim.x) w[i] = Wf[i];
    if (threadIdx.x < 10) bb[threadIdx.x] = bfv[threadIdx.x];
    __syncthreads();

    const int i = blockIdx.x * blockDim.x + threadIdx.x;
    if (i >= n) return;
    const float* hr = h + (long)i * 64;
    float logit[10];
#pragma unroll
    for (int o = 0; o < 10; ++o) logit[o] = bb[o];
    for (int k = 0; k < 64; ++k) {
        const float hv = hr[k];
#pragma unroll
        for (int o = 0; o < 10; ++o) logit[o] = fmaf(hv, w[k * 10 + o], logit[o]);
    }
    float m = logit[0];
#pragma unroll
    for (int o = 1; o < 10; ++o) m = fmaxf(m, logit[o]);
    float sum = 0.0f;
#pragma unroll
    for (int o = 0; o < 10; ++o) { logit[o] = __expf(logit[o] - m); sum += logit[o]; }
    const float inv = 1.0f / sum;
    float* orow = out + (long)i * 10;
#pragma unroll
    for (int o = 0; o < 10; ++o) orow[o] = logit[o] * inv;
}

extern "C" void kernel_launch(void* const* d_in, const int* in_sizes, int n_in,
                              void* d_out, int out_size, void* d_ws, size_t ws_size,
                              hipStream_t stream) {
    const float* x  = (const float*)d_in[0];
    const int*  ei  = (const int*)d_in[1];
    const float* W1 = (const float*)d_in[2];
    const float* b1 = (const float*)d_in[3];
    const float* W2 = (const float*)d_in[4];
    const float* b2 = (const float*)d_in[5];
    const float* Wf = (const float*)d_in[6];
    const float* bf = (const float*)d_in[7];
    float* out = (float*)d_out;

    const int N = in_sizes[0] / 128;       // 100000 nodes
    const int E = in_sizes[1] / 2;         // 1.6M edges
    const int* src = ei;                   // edge_index[0]
    const int* dst = ei + E;               // edge_index[1]

    // workspace layout (floats)
    float* wsf  = (float*)d_ws;
    const size_t nPad = ((size_t)N + 1023) & ~(size_t)1023;
    float* dis  = wsf;                         // [N]
    float* bufA = wsf + nPad;                  // [N*128]  t1s ; later t2s | h2
    float* bufB = bufA + (size_t)N * 128;      // [N*128]  h1 accumulator
    float* t2s  = bufA;                        // [N*64]
    float* h2   = bufA + (size_t)N * 64;       // [N*64]

    const int T = 256;

    // 1. symmetric norm:  dis = rsqrt(1 + in_degree)
    deg_init_kernel<<<(N + T - 1) / T, T, 0, stream>>>(dis, N);
    deg_count_kernel<<<(E + T - 1) / T, T, 0, stream>>>(dst, dis, E);
    make_dis_kernel<<<(N + T - 1) / T, T, 0, stream>>>(dis, N);

    // 2. layer 1:  t1s = dis * (x @ W1)   (WMMA);  h1 initialised with self-loop term
    gemm_rowscale_kernel<128, 128><<<(N + 127) / 128, T, 0, stream>>>(
        x, W1, dis, bufA, bufB, N);
    {
        const long tot = (long)E * (128 / 4);
        scatter_add_kernel<128><<<(int)((tot + T - 1) / T), T, 0, stream>>>(
            src, dst, bufA, bufB, E);
    }
    finish_layer_kernel<128><<<(int)(((long)N * 128 + T - 1) / T), T, 0, stream>>>(
        bufB, dis, b1, N);

    // 3. layer 2:  t2s = dis * (h1 @ W2)  (WMMA);  h2 initialised with self-loop term
    gemm_rowscale_kernel<128, 64><<<(N + 127) / 128, T, 0, stream>>>(
        bufB, W2, dis, t2s, h2, N);
    {
        const long tot = (long)E * (64 / 4);
        scatter_add_kernel<64><<<(int)((tot + T - 1) / T), T, 0, stream>>>(
            src, dst, t2s, h2, E);
    }
    finish_layer_kernel<64><<<(int)(((long)N * 64 + T - 1) / T), T, 0, stream>>>(
        h2, dis, b2, N);

    // 4. head + softmax
    head_softmax_kernel<<<(N + T - 1) / T, T, 0, stream>>>(h2, Wf, bf, out, N);
}